// NodeBlock_30391188586590
// MI455X (gfx1250) — compile-verified
//
#include <hip/hip_runtime.h>

// ---------------------------------------------------------------------------
// NodeBlock: proj = edge_attrs @ proj_W + proj_b            [E,64]
//            summed/counts = segment_sum(proj, dst)         [N,64],[N]
//            out = (summed / max(counts,1)) @ upd_W + upd_b [N,64]
// fp32 end-to-end using V_WMMA_F32_16X16X4_F32 (wave32, CDNA5), with
// double-buffered GLOBAL_LOAD_ASYNC_TO_LDS_B128 staging of edge tiles.
// ---------------------------------------------------------------------------

typedef float v2f __attribute__((ext_vector_type(2)));
typedef float v8f __attribute__((ext_vector_type(8)));

#define WAVES 8          // waves per 256-thread block
#define TPW   4          // 16-row tiles per wave
#define LDSTR 68         // padded LDS row stride (floats): bank = (4*row + k) % 64

__device__ __forceinline__ void atomAddF(float* p, float v) {
    __hip_atomic_fetch_add(p, v, __ATOMIC_RELAXED, __HIP_MEMORY_SCOPE_AGENT);
}

__device__ __forceinline__ unsigned ldsAddr(const void* p) {
    // LDS aperture keeps the wave-relative LDS offset in addr[31:0].
    return (unsigned)(uintptr_t)p;
}

__global__ void zero_ws(float* __restrict__ ws, int total) {
    int stride = gridDim.x * blockDim.x;
    for (int i = blockIdx.x * blockDim.x + threadIdx.x; i < total; i += stride)
        ws[i] = 0.0f;
}

// --------------------------- edge projection + scatter ----------------------
__global__ __launch_bounds__(256) void edge_proj_scatter(
    const float* __restrict__ edge_attrs,   // [E,64]
    const float* __restrict__ projW,        // [64,64] row-major (K,N)
    const float* __restrict__ projB,        // [64]
    const int*   __restrict__ dst,          // [E]
    float* __restrict__ summed,             // [N,64]
    float* __restrict__ counts,             // [N]
    int tiles)                              // E/16
{
    __shared__ float Wt[64 * LDSTR];               // Wt[n*LDSTR + k] = W[k][n]
    __shared__ float At[2][WAVES][16 * LDSTR];     // double-buffered edge tiles
    __shared__ int   Dn[WAVES][16];                // per-wave dst indices

    const int tid  = threadIdx.x;
    const int wave = tid >> 5;
    const int lane = tid & 31;
    const int m    = lane & 15;            // M (or N) index within fragment
    const int koff = (lane >> 4) * 2;      // K sub-offset for A/B fragments

    // Stage proj_W transposed into LDS (whole block cooperates).
    for (int i = tid; i < 64 * 64; i += 256) {
        int k = i >> 6, n = i & 63;
        Wt[n * LDSTR + k] = projW[i];
    }
    __syncthreads();

    // Per-lane bias values for the 4 column tiles.
    float pb[4];
#pragma unroll
    for (int nt = 0; nt < 4; ++nt) pb[nt] = projB[nt * 16 + m];

    // This lane's slice of the async tile copy: 2 x b128 per tile.
    const int ldRow = lane >> 2;           // rows 0..7  (x2 below)
    const int ldC4  = (lane & 3) * 16;     // float col 0/16/32/48

    const int tile0 = (blockIdx.x * WAVES + wave) * TPW;
    int rem = tiles - tile0;
    const int myTiles = rem < 0 ? 0 : (rem > TPW ? TPW : rem);

    // Issue one tile's worth of async global->LDS B128 copies (2 per wave).
    auto issueTile = [&](int tile, int buf) {
#pragma unroll
        for (int half = 0; half < 2; ++half) {
            const int row = ldRow + half * 8;
            unsigned loff = ldsAddr(&At[buf][wave][row * LDSTR + ldC4]);
            unsigned goff = ((unsigned)(tile * 16 + row) * 64 + ldC4) * 4u;
            asm volatile(
                "global_load_async_to_lds_b128 %0, %1, %2\n"
                :: "v"(loff), "v"(goff), "s"(edge_attrs)
                : "memory");
        }
    };

    if (myTiles > 0) issueTile(tile0, 0);

    for (int t = 0; t < myTiles; ++t) {
        const int tile = tile0 + t;
        const int buf  = t & 1;
        const int ebase = tile * 16;

        if (t + 1 < myTiles) {
            // WAR fence: prior ds_reads of buf^1 must have completed before
            // async writes may land there (DScnt and ASYNCcnt are unordered).
            asm volatile("s_wait_dscnt 0x0" ::: "memory");
            issueTile(tile + 1, buf ^ 1);
        }

        if (lane < 16) Dn[wave][lane] = dst[ebase + lane];

        // RAW fence: current tile's async group (2 ops) must be complete.
        if (t + 1 < myTiles)
            asm volatile("s_wait_asynccnt 0x8" ::: "memory");
        else
            asm volatile("s_wait_asynccnt 0x0" ::: "memory");
        __builtin_amdgcn_wave_barrier();

        const float* Aw = &At[buf][wave][0];
        v8f acc[4] = {v8f{0.f}, v8f{0.f}, v8f{0.f}, v8f{0.f}};
#pragma unroll
        for (int kk = 0; kk < 16; ++kk) {
            const int k0 = kk * 4 + koff;
            v2f a = *reinterpret_cast<const v2f*>(&Aw[m * LDSTR + k0]);
#pragma unroll
            for (int nt = 0; nt < 4; ++nt) {
                v2f b = *reinterpret_cast<const v2f*>(&Wt[(nt * 16 + m) * LDSTR + k0]);
                acc[nt] = __builtin_amdgcn_wmma_f32_16x16x4_f32(
                    false, a, false, b, (short)0, acc[nt], false, false);
            }
        }
        __builtin_amdgcn_wave_barrier();

        // Scatter (proj + bias) rows into summed[] via fp32 L2 atomics.
        // C layout: vgpr r, lanes 0-15 -> M=r, N=lane; lanes 16-31 -> M=r+8.
        const int mAdd = (lane >> 4) << 3;
#pragma unroll
        for (int nt = 0; nt < 4; ++nt) {
            const int n  = nt * 16 + m;
            const float bb = pb[nt];
#pragma unroll
            for (int r = 0; r < 8; ++r) {
                const int node = Dn[wave][r + mAdd];
                atomAddF(&summed[(size_t)node * 64 + n], acc[nt][r] + bb);
            }
        }
        if (lane < 16) atomAddF(&counts[Dn[wave][lane]], 1.0f);
        __builtin_amdgcn_wave_barrier();
    }
}

// ------------------------------- node update --------------------------------
__global__ __launch_bounds__(256) void node_update(
    const float* __restrict__ summed,       // [N,64]
    const float* __restrict__ counts,       // [N]
    const float* __restrict__ updW,         // [64,64]
    const float* __restrict__ updB,         // [64]
    float* __restrict__ out,                // [N,64]
    int tiles)                              // N/16
{
    __shared__ float Wt[64 * LDSTR];
    __shared__ float At[WAVES][16 * LDSTR];

    const int tid  = threadIdx.x;
    const int wave = tid >> 5;
    const int lane = tid & 31;
    const int m    = lane & 15;
    const int koff = (lane >> 4) * 2;

    for (int i = tid; i < 64 * 64; i += 256) {
        int k = i >> 6, n = i & 63;
        Wt[n * LDSTR + k] = updW[i];
    }
    __syncthreads();

    float ub[4];
#pragma unroll
    for (int nt = 0; nt < 4; ++nt) ub[nt] = updB[nt * 16 + m];

    const int tile0 = (blockIdx.x * WAVES + wave) * TPW;

    for (int t = 0; t < TPW; ++t) {
        const int tile = tile0 + t;
        if (tile >= tiles) break;
        const int nbase = tile * 16;

        // Stage mean-aggregated rows: summed / max(count,1).
#pragma unroll
        for (int it = 0; it < 8; ++it) {
            int idx = it * 32 + lane;
            int row = idx >> 4;
            int c4  = (idx & 15) * 4;
            const int node = nbase + row;
            const float inv = 1.0f / fmaxf(counts[node], 1.0f);
            float4 v = *reinterpret_cast<const float4*>(
                &summed[(size_t)node * 64 + c4]);
            v.x *= inv; v.y *= inv; v.z *= inv; v.w *= inv;
            *reinterpret_cast<float4*>(&At[wave][row * LDSTR + c4]) = v;
        }
        __builtin_amdgcn_wave_barrier();

        v8f acc[4] = {v8f{0.f}, v8f{0.f}, v8f{0.f}, v8f{0.f}};
#pragma unroll
        for (int kk = 0; kk < 16; ++kk) {
            const int k0 = kk * 4 + koff;
            v2f a = *reinterpret_cast<const v2f*>(&At[wave][m * LDSTR + k0]);
#pragma unroll
            for (int nt = 0; nt < 4; ++nt) {
                v2f b = *reinterpret_cast<const v2f*>(&Wt[(nt * 16 + m) * LDSTR + k0]);
                acc[nt] = __builtin_amdgcn_wmma_f32_16x16x4_f32(
                    false, a, false, b, (short)0, acc[nt], false, false);
            }
        }
        __builtin_amdgcn_wave_barrier();

        const int mAdd = (lane >> 4) << 3;
#pragma unroll
        for (int nt = 0; nt < 4; ++nt) {
            const int n = nt * 16 + m;
            const float bb = ub[nt];
#pragma unroll
            for (int r = 0; r < 8; ++r) {
                const int node = nbase + r + mAdd;
                out[(size_t)node * 64 + n] = acc[nt][r] + bb;
            }
        }
        __builtin_amdgcn_wave_barrier();
    }
}

// ------------------------------- launcher -----------------------------------
extern "C" void kernel_launch(void* const* d_in, const int* in_sizes, int n_in,
                              void* d_out, int out_size, void* d_ws, size_t ws_size,
                              hipStream_t stream) {
    const float* edge_attrs = (const float*)d_in[0];
    const float* projW      = (const float*)d_in[1];
    const float* projB      = (const float*)d_in[2];
    const float* updW       = (const float*)d_in[3];
    const float* updB       = (const float*)d_in[4];
    const int*   dst        = (const int*)d_in[5];

    const int E = in_sizes[0] / 64;
    const int N = out_size / 64;

    float* summed = (float*)d_ws;                 // [N,64]
    float* counts = summed + (size_t)N * 64;      // [N]
    float* out    = (float*)d_out;

    // 1) zero accumulators (atomics below must start from 0 every call)
    zero_ws<<<1024, 256, 0, stream>>>(summed, N * 65);

    // 2) edge projection + scatter-add
    const int tilesB  = E / 16;                   // E is a multiple of 16
    const int wavesB  = (tilesB + TPW - 1) / TPW;
    const int blocksB = (wavesB + WAVES - 1) / WAVES;
    edge_proj_scatter<<<blocksB, 256, 0, stream>>>(
        edge_attrs, projW, projB, dst, summed, counts, tilesB);

    // 3) mean + node update GEMM
    const int tilesC  = N / 16;                   // N is a multiple of 16
    const int wavesC  = (tilesC + TPW - 1) / TPW;
    const int blocksC = (wavesC + WAVES - 1) / WAVES;
    node_update<<<blocksC, 256, 0, stream>>>(
        summed, counts, updW, updB, out, tilesC);
}